// MultiHeadAttention_52802327937413
// MI455X (gfx1250) — compile-verified
//
#include <hip/hip_runtime.h>
#include <hip/hip_bf16.h>
#include <math.h>

typedef __attribute__((ext_vector_type(2))) float v2f;
typedef __attribute__((ext_vector_type(8))) float v8f;

#define NUM_HEAD 8
#define HIDDEN   1024
#define BB       2
#define QQ       256
#define KK       256
#define DHEAD    128   // HIDDEN / NUM_HEAD

// ---------------------------------------------------------------------------
// gfx1250 hardware tanh (V_TANH_F32, TRANS pipe). Fallback: exp2-based.
// ---------------------------------------------------------------------------
__device__ __forceinline__ float dev_tanh(float x)
{
#if __has_builtin(__builtin_amdgcn_tanhf)
    return __builtin_amdgcn_tanhf(x);
#else
    float e = __builtin_amdgcn_exp2f(x * 2.8853900817779268f);
    return (e - 1.0f) / (e + 1.0f);
#endif
}

// ---------------------------------------------------------------------------
// CDNA5 async global->LDS copy (ASYNCcnt-tracked), inline asm (portable
// across the two toolchains' builtin-arity skew).
// ---------------------------------------------------------------------------
__device__ __forceinline__ void async_ld_b128(unsigned lds_off, const float* gaddr)
{
    asm volatile("global_load_async_to_lds_b128 %0, %1, off"
                 :: "v"(lds_off), "v"(gaddr) : "memory");
}

__device__ __forceinline__ void wait_async_le1()
{
#if __has_builtin(__builtin_amdgcn_s_wait_asynccnt)
    __builtin_amdgcn_s_wait_asynccnt(1);
#else
    asm volatile("s_wait_asynccnt 0x1" ::: "memory");
#endif
}

__device__ __forceinline__ void wait_async_le0()
{
#if __has_builtin(__builtin_amdgcn_s_wait_asynccnt)
    __builtin_amdgcn_s_wait_asynccnt(0);
#else
    asm volatile("s_wait_asynccnt 0x0" ::: "memory");
#endif
}

// ---------------------------------------------------------------------------
// Wave-level fp32 WMMA: one wave computes a 16x64 tile of C = A @ B
// (row-major) via v_wmma_f32_16x16x4_f32. Layouts per ISA 7.12.2.
// ---------------------------------------------------------------------------
__device__ __forceinline__ void wmma_tile_16x64(const float* __restrict__ A, int lda,
                                                const float* __restrict__ Bm, int ldb,
                                                float* __restrict__ C, int ldc,
                                                int K, int tm, int tn, int lane)
{
    v8f acc0 = {}, acc1 = {}, acc2 = {}, acc3 = {};
    const int mrow  = tm + (lane & 15);
    const int col   = (lane & 15);
    const int khalf = (lane >> 4) << 1;     // 0 for lanes 0-15, 2 for lanes 16-31

    for (int k0 = 0; k0 < K; k0 += 4) {
        const int kk = k0 + khalf;
        v2f a;
        a.x = A[(size_t)mrow * lda + kk];
        a.y = A[(size_t)mrow * lda + kk + 1];

        const float* brow0 = Bm + (size_t)kk * ldb;
        const float* brow1 = Bm + (size_t)(kk + 1) * ldb;
        v2f b0, b1, b2, b3;
        b0.x = brow0[tn +  0 + col];  b0.y = brow1[tn +  0 + col];
        b1.x = brow0[tn + 16 + col];  b1.y = brow1[tn + 16 + col];
        b2.x = brow0[tn + 32 + col];  b2.y = brow1[tn + 32 + col];
        b3.x = brow0[tn + 48 + col];  b3.y = brow1[tn + 48 + col];

        acc0 = __builtin_amdgcn_wmma_f32_16x16x4_f32(false, a, false, b0, (short)0, acc0, false, false);
        acc1 = __builtin_amdgcn_wmma_f32_16x16x4_f32(false, a, false, b1, (short)0, acc1, false, false);
        acc2 = __builtin_amdgcn_wmma_f32_16x16x4_f32(false, a, false, b2, (short)0, acc2, false, false);
        acc3 = __builtin_amdgcn_wmma_f32_16x16x4_f32(false, a, false, b3, (short)0, acc3, false, false);
    }

    const int crow = tm + 8 * (lane >> 4);
    const int ccol = lane & 15;
#pragma unroll
    for (int r = 0; r < 8; ++r) {
        float* crp = C + (size_t)(crow + r) * ldc + tn;
        crp[ 0 + ccol] = acc0[r];
        crp[16 + ccol] = acc1[r];
        crp[32 + ccol] = acc2[r];
        crp[48 + ccol] = acc3[r];
    }
}

// ---------------------------------------------------------------------------
// Kernel 1: projection GEMM  C[M,N] = A[M,K] @ W[K,N]   (M=512, N=K=1024)
// ---------------------------------------------------------------------------
__global__ void __launch_bounds__(256)
proj_gemm_kernel(const float* __restrict__ A, const float* __restrict__ W,
                 float* __restrict__ C, int M, int N, int K)
{
    const int lane = threadIdx.x & 31;
    const int wave = threadIdx.x >> 5;
    const int tile = blockIdx.x * 8 + wave;
    const int tiles_n = N / 64;
    const int tm = (tile / tiles_n) * 16;
    const int tn = (tile % tiles_n) * 64;
    if (tm >= M) return;
    wmma_tile_16x64(A, K, W, N, C, N, K, tm, tn, lane);
}

// ---------------------------------------------------------------------------
// Kernel 2 (fused logits + mask + softmax):
//   logits[b,h,q,k] = sum_{d in head h} tanh(hq+hk)*Va  - penalty(b,k)
//   weights[b,h,q,:] = softmax_k(logits)
//
// Block = 256 threads = 8 waves; wave w == head w (wave32). Lane covers 4
// hidden dims (d = tid*4). Each block: one b, 2 q rows, all 256 k.
// hk rows double-buffered via global_load_async_to_lds_b128 (per-thread 16B
// slots -> in-order per-wave ASYNCcnt is the only sync needed).
// After the 5-step butterfly every lane holds the (q,k) sum; lane k%32 keeps
// it in register v[j][k/32] (v_cndmask, no divergent stores). Softmax then
// runs entirely in registers; only final weights are written (coalesced).
// ---------------------------------------------------------------------------
__global__ void __launch_bounds__(256)
logits_softmax_kernel(const float* __restrict__ hq, const float* __restrict__ hk,
                      const float* __restrict__ Va, const int* __restrict__ mask,
                      float* __restrict__ weights)
{
    __shared__ float kbuf[2][HIDDEN];         // 2 x 4KB double buffer

    const int t    = threadIdx.x;
    const int lane = t & 31;
    const int head = t >> 5;
    const int b    = blockIdx.x >> 7;         // Q/2 = 128 blocks per batch
    const int q0   = (blockIdx.x & 127) * 2;
    const int d    = t * 4;                   // == head*128 + lane*4

    const float4 va = *reinterpret_cast<const float4*>(Va + d);
    float4 hqv[2];
#pragma unroll
    for (int j = 0; j < 2; ++j)
        hqv[j] = *reinterpret_cast<const float4*>(hq + ((size_t)(b * QQ + q0 + j)) * HIDDEN + d);

    // penalty for the k this lane will own in chunk i: k = 32*i + lane
    float pen[8];
#pragma unroll
    for (int i = 0; i < 8; ++i)
        pen[i] = (1.0f - (float)mask[b * KK + 32 * i + lane]) * 99.0f;

    const float* hkbase = hk + (size_t)b * KK * HIDDEN + d;   // this thread's 16B column
    const unsigned lds_off[2] = {
        (unsigned)(uintptr_t)(&kbuf[0][d]),
        (unsigned)(uintptr_t)(&kbuf[1][d])
    };

    // prologue: stage row k=0
    async_ld_b128(lds_off[0], hkbase);

    float v[2][8];
#pragma unroll
    for (int i = 0; i < 8; ++i) { v[0][i] = 0.0f; v[1][i] = 0.0f; }

#pragma unroll
    for (int i = 0; i < 8; ++i) {             // constant index into v[][]
        for (int kk = 0; kk < 32; ++kk) {     // dynamic inner loop
            const int k = 32 * i + kk;
            if (k + 1 < KK) {
                async_ld_b128(lds_off[(k + 1) & 1], hkbase + (size_t)(k + 1) * HIDDEN);
                wait_async_le1();             // row k landed (in-order per wave)
            } else {
                wait_async_le0();
            }
            const float4 hkv = *reinterpret_cast<const float4*>(&kbuf[k & 1][d]);

#pragma unroll
            for (int j = 0; j < 2; ++j) {
                float s = dev_tanh(hqv[j].x + hkv.x) * va.x
                        + dev_tanh(hqv[j].y + hkv.y) * va.y
                        + dev_tanh(hqv[j].z + hkv.z) * va.z
                        + dev_tanh(hqv[j].w + hkv.w) * va.w;
#pragma unroll
                for (int off = 16; off > 0; off >>= 1)
                    s += __shfl_xor(s, off, 32);
                // lane kk keeps the value for k (all lanes hold the full sum)
                v[j][i] = (lane == kk) ? (s - pen[i]) : v[j][i];
            }
        }
    }

    // in-register softmax over k for each of this wave's 2 rows
#pragma unroll
    for (int j = 0; j < 2; ++j) {
        float m = -INFINITY;
#pragma unroll
        for (int i = 0; i < 8; ++i) m = fmaxf(m, v[j][i]);
#pragma unroll
        for (int off = 16; off > 0; off >>= 1)
            m = fmaxf(m, __shfl_xor(m, off, 32));

        float ssum = 0.0f;
#pragma unroll
        for (int i = 0; i < 8; ++i) {
            v[j][i] = __expf(v[j][i] - m);    // v_exp_f32
            ssum += v[j][i];
        }
#pragma unroll
        for (int off = 16; off > 0; off >>= 1)
            ssum += __shfl_xor(ssum, off, 32);

        const float inv = 1.0f / ssum;
        float* wrow = weights + (((size_t)(b * NUM_HEAD + head) * QQ) + (q0 + j)) * KK;
#pragma unroll
        for (int i = 0; i < 8; ++i)
            wrow[32 * i + lane] = v[j][i] * inv;
    }
}

// ---------------------------------------------------------------------------
// Kernel 3: ctx[b,q,h*128+d] = sum_k weights[b,h,q,k] * values[b,k,h*128+d]
// Per (b,h): [256x256]@[256x128] fp32 WMMA.
// ---------------------------------------------------------------------------
__global__ void __launch_bounds__(256)
ctx_gemm_kernel(const float* __restrict__ weights, const float* __restrict__ values,
                float* __restrict__ ctx)
{
    const int lane = threadIdx.x & 31;
    const int wave = threadIdx.x >> 5;
    const int bh   = blockIdx.y;
    const int b    = bh >> 3;
    const int h    = bh & 7;
    const int tile = blockIdx.x * 8 + wave;    // 0..31  (16 tm tiles x 2 tn tiles)
    const int tm   = (tile >> 1) * 16;
    const int tn   = (tile & 1) * 64;

    const float* A  = weights + (size_t)bh * QQ * KK;                 // [256,256], lda=256
    const float* Bm = values  + (size_t)b * KK * HIDDEN + h * DHEAD;  // [256,128], ldb=1024
    float*       C  = ctx     + (size_t)b * QQ * HIDDEN + h * DHEAD;  // [256,128], ldc=1024
    wmma_tile_16x64(A, KK, Bm, HIDDEN, C, HIDDEN, KK, tm, tn, lane);
}

// ---------------------------------------------------------------------------
extern "C" void kernel_launch(void* const* d_in, const int* in_sizes, int n_in,
                              void* d_out, int out_size, void* d_ws, size_t ws_size,
                              hipStream_t stream)
{
    (void)in_sizes; (void)n_in; (void)out_size; (void)ws_size;

    const float* query  = (const float*)d_in[0];   // [B,Q,H]
    const float* keys   = (const float*)d_in[1];   // [B,K,H]
    const float* values = (const float*)d_in[2];   // [B,K,H]
    const int*   mask   = (const int*)  d_in[3];   // [B,K]
    const float* W_q    = (const float*)d_in[4];   // [H,H]
    const float* W_k    = (const float*)d_in[5];   // [H,H]
    const float* V_a    = (const float*)d_in[6];   // [H]

    float* out     = (float*)d_out;
    float* ctx     = out;                                  // [B,Q,H]      = 524288
    float* weights = out + (size_t)BB * QQ * HIDDEN;       // [B,8,Q,K]    = 1048576

    float* hq = (float*)d_ws;                              // [B*Q, H] = 512x1024
    float* hk = hq + (size_t)BB * QQ * HIDDEN;             // [B*K, H] = 512x1024

    // 1) projections via fp32 WMMA:  (512x1024)@(1024x1024)
    proj_gemm_kernel<<<64, 256, 0, stream>>>(query, W_q, hq, BB * QQ, HIDDEN, HIDDEN);
    proj_gemm_kernel<<<64, 256, 0, stream>>>(keys,  W_k, hk, BB * KK, HIDDEN, HIDDEN);

    // 2) fused tanh-logits + mask + softmax -> final weights (256 blocks)
    logits_softmax_kernel<<<BB * (QQ / 2), 256, 0, stream>>>(hq, hk, V_a, mask, weights);

    // 3) context GEMM via fp32 WMMA
    ctx_gemm_kernel<<<dim3(4, BB * NUM_HEAD), 256, 0, stream>>>(weights, values, ctx);
}